// DSConv_63797444214798
// MI455X (gfx1250) — compile-verified
//
#include <hip/hip_runtime.h>
#include <math.h>

// ---------------------------------------------------------------------------
// DSConv forward for MI455X (gfx1250, wave32).
// Both convolutions run on V_WMMA_F32_16X16X4_F32 (f32 in/out).
// Memory-bound workload (~48 GFLOP vs ~1.2 GB traffic @ 23.3 TB/s):
//   P0 zero stat buckets
//   P1 3x3 offset conv (WMMA) -> offRaw[B,18,H,W] + bucketed sum/sumsq
//   P2 reduce buckets -> mean/invstd (18 ch)
//   P3 snake grid + bilinear sample (to LDS) + (1x9) conv (WMMA)
//      -> d_out (raw) + bucketed sum/sumsq
//   P4 reduce buckets -> mean/invstd (32 ch)
//   P5 BN + exact GELU, in place on d_out
// GEMM K-dim ordered k = tap*32 + ci (weights permuted at LDS fill); image
// tiles stored ci-innermost with pad-34 stride so each lane's B pair is one
// aligned, bank-conflict-free ds_load_b64 at an immediate offset: the
// unrolled WMMA loops need no per-step address VALU and no operand moves.
// offset_b is dropped: train-mode BN is invariant to a per-channel constant.
// ---------------------------------------------------------------------------

#define BB   16
#define CIN  32
#define HH   320
#define WW   320
#define COUT 32
#define KS   9
#define COFF 18
#define KDIM (CIN * KS)          // 288
#define NB   1024                // stat buckets (spread global atomics)
#define NTOT (BB * HH * WW)      // per-channel reduction count
#define BN_EPS 1e-5f
#define WTILES (WW / 64)         // 5 blocks per row
#define CST  34                  // ci-innermost stride (pad: banks + align)

typedef __attribute__((ext_vector_type(2))) float v2f;
typedef __attribute__((ext_vector_type(8))) float v8f;

__device__ __forceinline__ v8f wmma_f32_k4(v2f a, v2f b, v8f c) {
  // D = A(16x4 f32) * B(4x16 f32) + C(16x16 f32); 8-arg VOP3P form.
  return __builtin_amdgcn_wmma_f32_16x16x4_f32(
      false, a, false, b, (short)0, c, false, false);
}

// ---------------------------------------------------------------------------
__global__ void zero_kernel(float* __restrict__ p, int n) {
  int i = blockIdx.x * blockDim.x + threadIdx.x;
  if (i < n) p[i] = 0.0f;
}

// ---------------------------------------------------------------------------
// P1: 3x3 conv, 32 -> 18 channels (padded to 32 rows of zero weights).
// One block = 64 output pixels of one (b,h) row.
// 8 waves = {M half 0/1} x {N tile 0..3}.
__global__ __launch_bounds__(256) void offconv_kernel(
    const float* __restrict__ x, const float* __restrict__ wofs,
    float* __restrict__ offRaw, float* __restrict__ buckets) {
  __shared__ float wlds[32 * KDIM];      // [co][tap*32 + ci], rows 18..31 = 0
  __shared__ float xlds[(3 * 67) * CST]; // [dy*67+col][ci], col = w0-1..w0+64
  __shared__ float s_sum[32];
  __shared__ float s_sq[32];

  const int bi = blockIdx.x;
  const int wt = bi % WTILES;
  const int h  = (bi / WTILES) % HH;
  const int b  = bi / (WTILES * HH);
  const int w0 = wt * 64;
  const int tid = threadIdx.x;

  // weights: wofs is [18][32][3][3]; permute to k = tap*32 + ci ordering.
  for (int i = tid; i < 32 * KDIM; i += 256) {
    int co = i / KDIM, rem = i % KDIM;
    int r = rem >> 5, ci = rem & 31;
    wlds[i] = (co < COFF) ? wofs[co * KDIM + ci * 9 + r] : 0.0f;
  }
  // input tile with 3x3 halo, zero padded at image borders (ci innermost)
  for (int i = tid; i < CIN * 3 * 66; i += 256) {
    int ci = i / (3 * 66);
    int rem = i % (3 * 66);
    int dy = rem / 66, col = rem % 66;
    int hin = h - 1 + dy, win = w0 - 1 + col;
    float v = 0.0f;
    if (hin >= 0 && hin < HH && win >= 0 && win < WW)
      v = x[((b * CIN + ci) * HH + hin) * WW + win];
    xlds[(dy * 67 + col) * CST + ci] = v;
  }
  if (tid < 32) { s_sum[tid] = 0.0f; s_sq[tid] = 0.0f; }
  __syncthreads();

  const int lane  = tid & 31;
  const int wid   = tid >> 5;
  const int half  = wid & 1;          // which 16 output channels
  const int ntile = wid >> 1;         // which 16 pixels
  const int n     = lane & 15;
  const int khi   = (lane >> 4) * 2;  // lanes 16..31 hold K = k0+2, k0+3
  const int colb  = ntile * 16 + n;

  // Per-wave bases: all per-step offsets below are compile-time constants.
  const float* abase = &wlds[(half * 16 + (lane & 15)) * KDIM + khi];
  const float* bbase = &xlds[colb * CST + khi];   // ci = c0 + khi

  v8f acc = {};
#pragma unroll
  for (int kb = 0; kb < KDIM / 4; ++kb) {
    const int k4  = 4 * kb;
    const int tap = k4 >> 5;          // constant per unrolled step
    const int c0  = k4 & 31;          // constant per unrolled step (<= 28)
    const int dyc = tap / 3, dxc = tap - dyc * 3;   // constants
    v2f a  = *(const v2f*)(abase + k4);                        // kA, kA+1
    v2f bf = *(const v2f*)(bbase + (dyc * 67 + dxc) * CST + c0); // ci, ci+1
    acc = wmma_f32_k4(a, bf, acc);
  }

  const int wpix = w0 + ntile * 16 + n;
  const int mofs = (lane >> 4) * 8;   // C/D: VGPR r -> row r (lo) / r+8 (hi)
#pragma unroll
  for (int r = 0; r < 8; ++r) {
    int co = half * 16 + mofs + r;
    float v = acc[r];
    if (co < COFF) {
      offRaw[((b * COFF + co) * HH + h) * WW + wpix] = v;
      atomicAdd(&s_sum[co], v);
      atomicAdd(&s_sq[co], v * v);
    }
  }
  __syncthreads();
  if (tid < COFF) {
    int bucket = bi & (NB - 1);
    atomicAdd(&buckets[tid * NB + bucket], s_sum[tid]);
    atomicAdd(&buckets[(COFF + tid) * NB + bucket], s_sq[tid]);
  }
}

// ---------------------------------------------------------------------------
// P2/P4: reduce NB buckets -> per-channel {mean, invstd}.
__global__ void stats_kernel(const float* __restrict__ buckets,
                             float* __restrict__ stats, int nch) {
  int ch = blockIdx.x * blockDim.x + threadIdx.x;
  if (ch >= nch) return;
  float s = 0.0f, q = 0.0f;
  for (int i = 0; i < NB; ++i) {
    s += buckets[ch * NB + i];
    q += buckets[(nch + ch) * NB + i];
  }
  float mean = s / (float)NTOT;
  float var  = q / (float)NTOT - mean * mean;   // biased var (jnp.var)
  stats[2 * ch]     = mean;
  stats[2 * ch + 1] = rsqrtf(var + BN_EPS);
}

// ---------------------------------------------------------------------------
// P3: snake grid + bilinear sample into LDS x_def halo tile, then (1x9) conv
// as WMMA GEMM. Raw conv output goes to d_out; stats bucketed.
__global__ __launch_bounds__(256) void main_kernel(
    const float* __restrict__ x, const float* __restrict__ wconv,
    const float* __restrict__ offRaw, const float* __restrict__ statsOff,
    const float* __restrict__ gOff, const float* __restrict__ bOff,
    float* __restrict__ convRaw, float* __restrict__ buckets) {
  __shared__ float wlds[COUT * KDIM];   // [co][tap*32 + ci]
  __shared__ float xdlds[72 * CST];     // [col][ci], halo pixels w0-4..w0+67
  __shared__ float s_px[72], s_py[72], s_val[72];
  __shared__ float s_sum[32], s_sq[32];

  const int bi = blockIdx.x;
  const int wt = bi % WTILES;
  const int h  = (bi / WTILES) % HH;
  const int b  = bi / (WTILES * HH);
  const int w0 = wt * 64;
  const int tid = threadIdx.x;

  // conv_w is [32][32][1][9]; permute to k = tap*32 + ci ordering.
  for (int i = tid; i < COUT * KDIM; i += 256) {
    int co = i / KDIM, rem = i % KDIM;
    int t = rem >> 5, ci = rem & 31;
    wlds[i] = wconv[co * KDIM + ci * 9 + t];
  }
  if (tid < 32) { s_sum[tid] = 0.0f; s_sq[tid] = 0.0f; }

  // Step 1: per-halo-pixel sampling coordinates from BN+tanh'd offsets.
  if (tid < 72) {
    int win = w0 - 4 + tid;
    if (win >= 0 && win < WW) {
      float sx = 0.0f, sy = 0.0f;
      for (int ch = 0; ch < KS; ++ch) {
        float v = offRaw[((b * COFF + ch) * HH + h) * WW + win];
        sx += tanhf((v - statsOff[2 * ch]) * statsOff[2 * ch + 1] * gOff[ch] +
                    bOff[ch]);
      }
      for (int ch = KS; ch < 2 * KS; ++ch) {
        float v = offRaw[((b * COFF + ch) * HH + h) * WW + win];
        sy += tanhf((v - statsOff[2 * ch]) * statsOff[2 * ch + 1] * gOff[ch] +
                    bOff[ch]);
      }
      float gx = -1.0f + 2.0f * win / (WW - 1) + (sx / 9.0f) * (2.0f / WW);
      float gy = -1.0f + 2.0f * h   / (HH - 1) + (sy / 9.0f) * (2.0f / HH);
      s_px[tid]  = (gx + 1.0f) * 0.5f * (WW - 1);  // align_corners=True
      s_py[tid]  = (gy + 1.0f) * 0.5f * (HH - 1);
      s_val[tid] = 1.0f;
    } else {                      // (1,K) conv zero padding along W
      s_px[tid] = 0.0f; s_py[tid] = 0.0f; s_val[tid] = 0.0f;
    }
  }
  __syncthreads();

  // Step 2: bilinear gather (zeros padding) -> x_def halo tile in LDS.
  for (int t = tid; t < CIN * 72; t += 256) {
    int i = t % 72, ci = t / 72;
    float out = 0.0f;
    if (s_val[i] != 0.0f) {
      float px = s_px[i], py = s_py[i];
      float x0f = floorf(px), y0f = floorf(py);
      float wx1 = px - x0f, wy1 = py - y0f;
      float wx0 = 1.0f - wx1, wy0 = 1.0f - wy1;
      int ix0 = (int)x0f, iy0 = (int)y0f;
      const float* base = x + ((size_t)(b * CIN + ci)) * (HH * WW);
      float v00 = 0.0f, v01 = 0.0f, v10 = 0.0f, v11 = 0.0f;
      if (iy0 >= 0 && iy0 < HH) {
        if (ix0 >= 0 && ix0 < WW)         v00 = base[iy0 * WW + ix0];
        if (ix0 + 1 >= 0 && ix0 + 1 < WW) v01 = base[iy0 * WW + ix0 + 1];
      }
      if (iy0 + 1 >= 0 && iy0 + 1 < HH) {
        if (ix0 >= 0 && ix0 < WW)         v10 = base[(iy0 + 1) * WW + ix0];
        if (ix0 + 1 >= 0 && ix0 + 1 < WW) v11 = base[(iy0 + 1) * WW + ix0 + 1];
      }
      out = wy0 * (wx0 * v00 + wx1 * v01) + wy1 * (wx0 * v10 + wx1 * v11);
    }
    xdlds[i * CST + ci] = out;   // ci innermost
  }
  __syncthreads();

  // Step 3: (1x9) conv GEMM: M=32 (2 halves), K=288, N=16 pixels per wave.
  const int lane  = tid & 31;
  const int wid   = tid >> 5;
  const int half  = wid & 1;
  const int ntile = wid >> 1;
  const int n     = lane & 15;
  const int khi   = (lane >> 4) * 2;
  const int colb  = ntile * 16 + n;

  const float* abase = &wlds[(half * 16 + (lane & 15)) * KDIM + khi];
  const float* bbase = &xdlds[colb * CST + khi];   // ci = c0 + khi

  v8f acc = {};
#pragma unroll
  for (int kb = 0; kb < KDIM / 4; ++kb) {
    const int k4  = 4 * kb;
    const int tap = k4 >> 5;          // constant per unrolled step
    const int c0  = k4 & 31;          // constant per unrolled step (<= 28)
    v2f a  = *(const v2f*)(abase + k4);
    v2f bf = *(const v2f*)(bbase + tap * CST + c0);   // ci, ci+1 contiguous
    acc = wmma_f32_k4(a, bf, acc);
  }

  const int wpix = w0 + ntile * 16 + n;
  const int mofs = (lane >> 4) * 8;
#pragma unroll
  for (int r = 0; r < 8; ++r) {
    int co = half * 16 + mofs + r;
    float v = acc[r];
    convRaw[((b * COUT + co) * HH + h) * WW + wpix] = v;
    atomicAdd(&s_sum[co], v);
    atomicAdd(&s_sq[co], v * v);
  }
  __syncthreads();
  if (tid < COUT) {
    int bucket = bi & (NB - 1);
    atomicAdd(&buckets[tid * NB + bucket], s_sum[tid]);
    atomicAdd(&buckets[(COUT + tid) * NB + bucket], s_sq[tid]);
  }
}

// ---------------------------------------------------------------------------
// P5: BN + exact GELU, in place on d_out.
__global__ void bngelu_kernel(float* __restrict__ data,
                              const float* __restrict__ stats,
                              const float* __restrict__ gamma,
                              const float* __restrict__ beta) {
  size_t i = (size_t)blockIdx.x * blockDim.x + threadIdx.x;
  if (i >= (size_t)BB * COUT * HH * WW) return;
  int c = (int)((i / (size_t)(HH * WW)) % COUT);
  float v = data[i];
  float y = (v - stats[2 * c]) * stats[2 * c + 1] * gamma[c] + beta[c];
  data[i] = 0.5f * y * (1.0f + erff(y * 0.70710678118654752f));
}

// ---------------------------------------------------------------------------
extern "C" void kernel_launch(void* const* d_in, const int* in_sizes, int n_in,
                              void* d_out, int out_size, void* d_ws,
                              size_t ws_size, hipStream_t stream) {
  const float* x            = (const float*)d_in[0];
  const float* offset_w     = (const float*)d_in[1];
  // d_in[2] = offset_b: mathematically a no-op under train-mode BN.
  const float* bn_off_gamma = (const float*)d_in[3];
  const float* bn_off_beta  = (const float*)d_in[4];
  const float* conv_w       = (const float*)d_in[5];
  const float* bn_gamma     = (const float*)d_in[6];
  const float* bn_beta      = (const float*)d_in[7];
  float* out = (float*)d_out;   // doubles as raw main-conv buffer
  float* ws  = (float*)d_ws;

  float* offRaw      = ws;                                  // B*18*H*W
  float* bucketsOff  = offRaw + (size_t)BB * COFF * HH * WW;
  float* bucketsConv = bucketsOff + 2 * COFF * NB;
  float* statsOff    = bucketsConv + 2 * COUT * NB;
  float* statsConv   = statsOff + 2 * COFF;

  const int nBuckets = 2 * COFF * NB + 2 * COUT * NB;
  zero_kernel<<<(nBuckets + 255) / 256, 256, 0, stream>>>(bucketsOff, nBuckets);

  const int nblocks = BB * HH * WTILES;   // 25600
  offconv_kernel<<<nblocks, 256, 0, stream>>>(x, offset_w, offRaw, bucketsOff);
  stats_kernel<<<1, 64, 0, stream>>>(bucketsOff, statsOff, COFF);
  main_kernel<<<nblocks, 256, 0, stream>>>(x, conv_w, offRaw, statsOff,
                                           bn_off_gamma, bn_off_beta, out,
                                           bucketsConv);
  stats_kernel<<<1, 64, 0, stream>>>(bucketsConv, statsConv, COUT);
  const size_t ntot = (size_t)BB * COUT * HH * WW;
  bngelu_kernel<<<(int)((ntot + 255) / 256), 256, 0, stream>>>(
      out, statsConv, bn_gamma, bn_beta);
}